// BasicRecurrentEntityEncoder_25494925869200
// MI455X (gfx1250) — compile-verified
//
#include <hip/hip_runtime.h>

// Problem dims (fixed by the reference)
#define BDIM 64
#define SDIM 64
#define LDIM 20
#define KDIM 20
#define DDIM 128
#define EPSV 1e-12f
#define PAD  4          // LDS row padding (floats) to avoid 64-bank conflicts

typedef __attribute__((ext_vector_type(2))) float v2f;
typedef __attribute__((ext_vector_type(8))) float v8f;

// ---------------------------------------------------------------------------
// Phase 1: encoded_sents[b,s,d] = sum_l mask[b,s,l] * E[prgrph[b,s,l], d] * pos[l,d]
// One block per (b,s), 128 threads (one per d). Gather-bound.
// ---------------------------------------------------------------------------
__global__ __launch_bounds__(128) void encode_kernel(
    const int* __restrict__ prgrph, const unsigned char* __restrict__ pmask,
    const float* __restrict__ emb, const float* __restrict__ pos,
    float* __restrict__ enc)
{
    __shared__ int   toks[LDIM];
    __shared__ float msk[LDIM];
    const int bs = blockIdx.x;
    const int d  = threadIdx.x;
    if (d < LDIM) {
        toks[d] = prgrph[bs * LDIM + d];
        msk[d]  = pmask[bs * LDIM + d] ? 1.0f : 0.0f;
    }
    __syncthreads();
    float acc = 0.0f;
    #pragma unroll
    for (int l = 0; l < LDIM; ++l) {
        if (msk[l] != 0.0f)           // uniform across block
            acc += emb[(long)toks[l] * DDIM + d] * pos[l * DDIM + d];
    }
    enc[(long)bs * DDIM + d] = acc;
}

// ---------------------------------------------------------------------------
// Phase 2: C(M,128) = A(M,128) @ B(128,128), fp32 WMMA 16x16x4.
// One wave per 16x16 output tile; grid = (M/16)*8.
// A-frag layout (ISA 7.12.2): lane<16 -> {K0,K1}, lane>=16 -> {K2,K3}, M=lane&15.
// B-frag mirrors: VGPR0 = row K0/K2 striped across lanes, VGPR1 = K1/K3.
// C/D: VGPR r -> M = r + 8*(lane>>4), N = lane&15.
// ---------------------------------------------------------------------------
__global__ __launch_bounds__(32) void gemm128_kernel(
    const float* __restrict__ A, const float* __restrict__ Bm,
    float* __restrict__ C)
{
    const int tile = blockIdx.x;
    const int mt   = tile >> 3;     // 8 N-tiles per M-tile (N = 128)
    const int nt   = tile & 7;
    const int lane = threadIdx.x;
    const int half = lane >> 4;
    const int l16  = lane & 15;

    const float* Arow = A + (long)(mt * 16 + l16) * DDIM;
    const float* Bcol = Bm + nt * 16 + l16;
    v8f acc = {};
    #pragma unroll 4
    for (int kb = 0; kb < 32; ++kb) {
        const int k0 = kb * 4 + 2 * half;
        v2f a, b;
        a.x = Arow[k0];
        a.y = Arow[k0 + 1];
        b.x = Bcol[(long)k0 * DDIM];
        b.y = Bcol[(long)(k0 + 1) * DDIM];
        acc = __builtin_amdgcn_wmma_f32_16x16x4_f32(
                  false, a, false, b, (short)0, acc, false, false);
    }
    #pragma unroll
    for (int r = 0; r < 8; ++r) {
        const int row = mt * 16 + r + 8 * half;
        C[(long)row * DDIM + nt * 16 + l16] = acc[r];
    }
}

// ---------------------------------------------------------------------------
// Phase 3: the recurrence. One block per batch b; h[b] (padded to 32 rows)
// lives in LDS for all 64 steps. Per step:
//   gate[k] = sigmoid(e . (h[k]+keys[k]))                 (wave shuffles)
//   hU      = h(32x128) @ U(128x128)                      (fp32 WMMA)
//   pre     = h + gate * relu(hU + keysV + eW)            (epilogue)
//   h       = sent_mask ? l2norm(pre) : h                 (row reduce + commit)
// 8 waves: wave w owns N-tile w (cols 16w..16w+15), M-tiles 0 and 1.
// ---------------------------------------------------------------------------
__global__ __launch_bounds__(256) void scan_kernel(
    const float* __restrict__ keys, const float* __restrict__ keysV,
    const float* __restrict__ enc,  const float* __restrict__ ew,
    const unsigned char* __restrict__ pmask, const float* __restrict__ U,
    float* __restrict__ out)
{
    __shared__ float h_s  [32][DDIM + PAD];
    __shared__ float pre_s[32][DDIM + PAD];
    __shared__ float ks_s [KDIM][DDIM + PAD];
    __shared__ float kv_s [KDIM][DDIM + PAD];
    __shared__ float e_s[DDIM], ew_s[DDIM];
    __shared__ float gate_s[KDIM], rn_s[KDIM];

    const int b    = blockIdx.x;
    const int tid  = threadIdx.x;
    const int wave = tid >> 5;
    const int lane = tid & 31;
    const int half = lane >> 4;
    const int l16  = lane & 15;
    const int ncol = wave * 16 + l16;   // this wave's output column

    for (int i = tid; i < 32 * DDIM; i += 256)
        h_s[i >> 7][i & 127] = 0.0f;
    for (int i = tid; i < KDIM * DDIM; i += 256) {
        ks_s[i >> 7][i & 127] = keys [(long)b * KDIM * DDIM + i];
        kv_s[i >> 7][i & 127] = keysV[(long)b * KDIM * DDIM + i];
    }
    __syncthreads();

    for (int s = 0; s < SDIM; ++s) {
        if (tid < DDIM) {
            e_s [tid] = enc[((long)b * SDIM + s) * DDIM + tid];
            ew_s[tid] = ew [((long)b * SDIM + s) * DDIM + tid];
        }
        __syncthreads();

        // ---- gate[k] = sigmoid(e . (h[k] + keys[k])) -----------------------
        for (int k = wave; k < KDIM; k += 8) {
            float p = 0.0f;
            #pragma unroll
            for (int d = lane; d < DDIM; d += 32)
                p += e_s[d] * (h_s[k][d] + ks_s[k][d]);
            #pragma unroll
            for (int off = 16; off > 0; off >>= 1)
                p += __shfl_xor(p, off, 32);
            if (lane == 0)
                gate_s[k] = 1.0f / (1.0f + __expf(-p));
        }
        __syncthreads();

        // ---- hU = h @ U via V_WMMA_F32_16X16X4_F32 -------------------------
        v8f acc0 = {}, acc1 = {};
        for (int kb = 0; kb < 32; ++kb) {
            const int k0 = kb * 4 + 2 * half;
            v2f a0, a1, bf;
            a0.x = h_s[l16][k0];          a0.y = h_s[l16][k0 + 1];
            a1.x = h_s[16 + l16][k0];     a1.y = h_s[16 + l16][k0 + 1];
            bf.x = U[(long)k0 * DDIM + ncol];
            bf.y = U[(long)(k0 + 1) * DDIM + ncol];
            acc0 = __builtin_amdgcn_wmma_f32_16x16x4_f32(
                       false, a0, false, bf, (short)0, acc0, false, false);
            acc1 = __builtin_amdgcn_wmma_f32_16x16x4_f32(
                       false, a1, false, bf, (short)0, acc1, false, false);
        }

        // ---- epilogue: pre = h + gate * relu(hU + keysV + eW) --------------
        #pragma unroll
        for (int r = 0; r < 8; ++r) {
            const int row0 = r + 8 * half;
            const int row1 = 16 + row0;
            float kv0 = (row0 < KDIM) ? kv_s[row0][ncol] : 0.0f;
            float g0  = (row0 < KDIM) ? gate_s[row0]     : 0.0f;
            float t0  = acc0[r] + kv0 + ew_s[ncol];
            t0 = t0 > 0.0f ? t0 : 0.0f;
            pre_s[row0][ncol] = h_s[row0][ncol] + g0 * t0;

            float kv1 = (row1 < KDIM) ? kv_s[row1][ncol] : 0.0f;
            float g1  = (row1 < KDIM) ? gate_s[row1]     : 0.0f;
            float t1  = acc1[r] + kv1 + ew_s[ncol];
            t1 = t1 > 0.0f ? t1 : 0.0f;
            pre_s[row1][ncol] = h_s[row1][ncol] + g1 * t1;
        }
        __syncthreads();

        // ---- per-row L2 norm ----------------------------------------------
        for (int k = wave; k < KDIM; k += 8) {
            float p = 0.0f;
            #pragma unroll
            for (int d = lane; d < DDIM; d += 32) {
                const float v = pre_s[k][d];
                p += v * v;
            }
            #pragma unroll
            for (int off = 16; off > 0; off >>= 1)
                p += __shfl_xor(p, off, 32);
            if (lane == 0)
                rn_s[k] = rsqrtf(fmaxf(p, EPSV));
        }
        __syncthreads();

        // ---- commit (sentence mask is uniform across the block) ------------
        if (pmask[((long)b * SDIM + s) * LDIM]) {
            for (int i = tid; i < KDIM * DDIM; i += 256)
                h_s[i >> 7][i & 127] = pre_s[i >> 7][i & 127] * rn_s[i >> 7];
        }
        __syncthreads();
    }

    for (int i = tid; i < KDIM * DDIM; i += 256)
        out[(long)b * KDIM * DDIM + i] = h_s[i >> 7][i & 127];
}

// ---------------------------------------------------------------------------
extern "C" void kernel_launch(void* const* d_in, const int* in_sizes, int n_in,
                              void* d_out, int out_size, void* d_ws, size_t ws_size,
                              hipStream_t stream) {
    (void)in_sizes; (void)n_in; (void)out_size; (void)ws_size;
    const int*           prgrph = (const int*)d_in[0];
    const unsigned char* pmask  = (const unsigned char*)d_in[1];
    const float*         keys   = (const float*)d_in[2];
    const float*         emb    = (const float*)d_in[3];
    const float*         pos    = (const float*)d_in[4];
    const float*         U      = (const float*)d_in[5];
    const float*         V      = (const float*)d_in[6];
    const float*         W      = (const float*)d_in[7];
    float* out = (float*)d_out;

    float* enc   = (float*)d_ws;                          // B*S*D
    float* ewbuf = enc   + (size_t)BDIM * SDIM * DDIM;    // B*S*D
    float* keysV = ewbuf + (size_t)BDIM * SDIM * DDIM;    // B*K*D

    encode_kernel<<<BDIM * SDIM, DDIM, 0, stream>>>(prgrph, pmask, emb, pos, enc);
    gemm128_kernel<<<(BDIM * KDIM / 16) * 8, 32, 0, stream>>>(keys, V, keysV);   // keys @ V (hoisted)
    gemm128_kernel<<<(BDIM * SDIM / 16) * 8, 32, 0, stream>>>(enc,  W, ewbuf);   // enc  @ W (batched)
    scan_kernel<<<BDIM, 256, 0, stream>>>(keys, keysV, enc, ewbuf, pmask, U, out);
}